// GraphConvolution_71511205478886
// MI455X (gfx1250) — compile-verified
//
#include <hip/hip_runtime.h>
#include <cstdint>
#include <cstddef>

// ---------------------------------------------------------------------------
// GraphConvolution SpMM (COO) for MI455X / gfx1250.
//
// out[r,:] = sum_e (rows[e]==r) vals[e] * annotations[cols[e],:]
//
// CSR-bin edges by destination row, then compute each 16-row output window as
// D = A x B on the WMMA pipe (V_WMMA_F32_16X16X4_F32):
//   A (16xK): one nonzero per column, A[row(k)-base, k] = val[k]
//   B (Kx16): B[k,:] = annotations[col[k], :]
// Output rows are written densely exactly once (no atomics on the 20 MB
// output).  Inner loop is branch-free and two-stage software-pipelined:
// metadata is fetched two 4-edge chunks ahead, the B fragment one chunk
// ahead, so the s_wait before each WMMA targets loads issued a full
// iteration earlier (gather latency hidden behind the matrix op).
// ---------------------------------------------------------------------------

typedef __attribute__((ext_vector_type(2))) float v2f;
typedef __attribute__((ext_vector_type(4))) float f4;
typedef __attribute__((ext_vector_type(8))) float v8f;

#define DFEAT 128
#define SCAN_THREADS 1024
#define CSR_PAD 16

// ---------------- helpers ----------------

__global__ void zero_i32_kernel(int* __restrict__ p, int n) {
  int i = blockIdx.x * blockDim.x + threadIdx.x;
  int s = gridDim.x * blockDim.x;
  for (; i < n; i += s) p[i] = 0;
}

__global__ void zero_f32_kernel(float* __restrict__ p, int n) {
  int i = blockIdx.x * blockDim.x + threadIdx.x;
  int s = gridDim.x * blockDim.x;
  for (; i < n; i += s) p[i] = 0.0f;
}

// Zero the CSR over-read pad (col=0 -> safe gather, val=0 -> a=0, row=-1).
__global__ void pad_kernel(int* __restrict__ csrCol, int* __restrict__ csrRow,
                           float* __restrict__ csrVal, int E) {
  int i = threadIdx.x;
  if (i < CSR_PAD) {
    csrCol[E + i] = 0;
    csrRow[E + i] = -1;
    csrVal[E + i] = 0.0f;
  }
}

// ---------------- CSR build: histogram -> scan -> fill ----------------

__global__ void hist_kernel(const int* __restrict__ rows,
                            int* __restrict__ counts, int E) {
  int i = blockIdx.x * blockDim.x + threadIdx.x;
  int s = gridDim.x * blockDim.x;
  for (; i < E; i += s) atomicAdd(&counts[rows[i]], 1);
}

// Single-block exclusive scan with running carry (n = 40000 -> 40 tiles).
__global__ __launch_bounds__(SCAN_THREADS)
void scan_kernel(const int* __restrict__ counts, int* __restrict__ rowStart,
                 int n) {
  __shared__ int sm[SCAN_THREADS];
  int carry = 0;
  for (int base = 0; base < n; base += SCAN_THREADS) {
    int i = base + (int)threadIdx.x;
    int x = (i < n) ? counts[i] : 0;
    sm[threadIdx.x] = x;
    __syncthreads();
    for (int off = 1; off < SCAN_THREADS; off <<= 1) {
      int t = (threadIdx.x >= (unsigned)off) ? sm[threadIdx.x - off] : 0;
      __syncthreads();
      sm[threadIdx.x] += t;
      __syncthreads();
    }
    int incl = sm[threadIdx.x];
    if (i < n) rowStart[i] = carry + (incl - x);
    carry += sm[SCAN_THREADS - 1];  // uniform across block
    __syncthreads();
  }
  if (threadIdx.x == 0) rowStart[n] = carry;
}

__global__ void fill_kernel(const int* __restrict__ rows,
                            const int* __restrict__ cols,
                            const float* __restrict__ vals,
                            const int* __restrict__ rowStart,
                            int* __restrict__ cursor,
                            int* __restrict__ csrCol, int* __restrict__ csrRow,
                            float* __restrict__ csrVal, int E) {
  int i = blockIdx.x * blockDim.x + threadIdx.x;
  int s = gridDim.x * blockDim.x;
  for (; i < E; i += s) {
    int r = rows[i];
    int pos = rowStart[r] + atomicAdd(&cursor[r], 1);
    csrCol[pos] = cols[i];
    csrRow[pos] = r;
    csrVal[pos] = vals[i];
  }
}

// ---------------- WMMA segment-sum SpMM ----------------
// One block per 16-row window; 8 waves, wave w owns feature cols [16w,16w+16).
// A fragment layout (32-bit A 16x4): lane L<16 -> M=L, VGPR pair = K{0,1};
// lanes 16-31 -> K{2,3}.  B (4x16): v0 = K0|K2, v1 = K1|K3 across lane halves.
// C/D: VGPR v <-> M = v + 8*(lane>=16), N = lane&15.

struct Meta {
  int2 c;
  int2 r;
  float2 v;
};

__device__ __forceinline__ Meta load_meta(const int* __restrict__ csrCol,
                                          const int* __restrict__ csrRow,
                                          const float* __restrict__ csrVal,
                                          int ka) {
  Meta mt;
  mt.c = *(const int2*)(csrCol + ka);    // ka is even -> 8B aligned
  mt.r = *(const int2*)(csrRow + ka);
  mt.v = *(const float2*)(csrVal + ka);
  return mt;
}

__global__ __launch_bounds__(256)
void wmma_spmm_kernel(const int* __restrict__ rowStart,
                      const int* __restrict__ csrCol,
                      const int* __restrict__ csrRow,
                      const float* __restrict__ csrVal,
                      const float* __restrict__ ann, float* __restrict__ out) {
  const int base = blockIdx.x * 16;  // first output row of window
  const int kBeg = rowStart[base];
  const int kEnd = rowStart[base + 16];
  const int lane = (int)(threadIdx.x & 31);
  const int wv = (int)(threadIdx.x >> 5);  // feature tile 0..7
  const int half = lane >> 4;              // 0: K{0,1}/M 0-7, 1: K{2,3}/M 8-15
  const int jbn = wv * 16 + (lane & 15);   // this lane's feature column
  const int rowm = base + (lane & 15);     // full row id this lane's A-row is

  v8f acc = {0.f, 0.f, 0.f, 0.f, 0.f, 0.f, 0.f, 0.f};

  // 4-aligned start: leading foreign entries fail the full-row compare -> a=0.
  int k0 = kBeg & ~3;
  int ka = k0 + 2 * half;  // even -> aligned b64 metadata loads

  if (k0 < kEnd) {
    // Pipeline prologue: metadata two chunks deep, B fragment one chunk deep.
    Meta m0 = load_meta(csrCol, csrRow, csrVal, ka);
    Meta m1 = load_meta(csrCol, csrRow, csrVal, ka + 4);
    v2f b0;
    b0.x = ann[m0.c.x * DFEAT + jbn];
    b0.y = ann[m0.c.y * DFEAT + jbn];

#pragma unroll 2
    for (; k0 < kEnd; k0 += 4, ka += 4) {
      Meta m2 = load_meta(csrCol, csrRow, csrVal, ka + 8);  // 2 chunks ahead

      v2f b1;  // 1 chunk ahead (meta issued a full iteration ago)
      b1.x = ann[m1.c.x * DFEAT + jbn];
      b1.y = ann[m1.c.y * DFEAT + jbn];

      v2f a;
      a.x = (m0.r.x == rowm) ? m0.v.x : 0.0f;
      a.y = (m0.r.y == rowm) ? m0.v.y : 0.0f;

      // b0's gathers were issued a full iteration ago -> latency hidden.
      acc = __builtin_amdgcn_wmma_f32_16x16x4_f32(
          /*neg_a=*/false, a, /*neg_b=*/false, b0,
          /*c_mod=*/(short)0, acc, /*reuse_a=*/false, /*reuse_b=*/false);

      m0 = m1;
      m1 = m2;
      b0 = b1;
    }
  }

  // Dense store: row = base + v + 8*half, col = jbn.
  float* o = out + (size_t)(base + 8 * half) * DFEAT + jbn;
#pragma unroll
  for (int v = 0; v < 8; ++v) o[(size_t)v * DFEAT] = acc[v];
}

// ---------------- fallback: wave-per-edge atomic scatter ----------------

__global__ __launch_bounds__(256)
void scatter_kernel(const int* __restrict__ rows, const int* __restrict__ cols,
                    const float* __restrict__ vals,
                    const float* __restrict__ ann, float* __restrict__ out,
                    int E) {
  const int lane = (int)(threadIdx.x & 31);
  int wid = (int)((blockIdx.x * blockDim.x + threadIdx.x) >> 5);
  const int nw = (int)((gridDim.x * blockDim.x) >> 5);
  for (int e = wid; e < E; e += nw) {
    const int r = rows[e];
    const int c = cols[e];
    const float v = vals[e];
    const f4 x = *(const f4*)(ann + (size_t)c * DFEAT + lane * 4);
    float* o = out + (size_t)r * DFEAT + lane * 4;
    unsafeAtomicAdd(o + 0, v * x.x);
    unsafeAtomicAdd(o + 1, v * x.y);
    unsafeAtomicAdd(o + 2, v * x.z);
    unsafeAtomicAdd(o + 3, v * x.w);
  }
}

// ---------------- launcher ----------------

extern "C" void kernel_launch(void* const* d_in, const int* in_sizes, int n_in,
                              void* d_out, int out_size, void* d_ws,
                              size_t ws_size, hipStream_t stream) {
  const int* rows = (const int*)d_in[0];
  const int* cols = (const int*)d_in[1];
  const float* vals = (const float*)d_in[2];
  const float* ann = (const float*)d_in[3];
  float* out = (float*)d_out;

  const int E = in_sizes[0];
  const int N = out_size / DFEAT;

  // counts, cursor, rowStart, csrCol, csrRow, csrVal (each csr: E + pad)
  const size_t need = (size_t)(3 * N + 1) * sizeof(int) +
                      (size_t)(E + CSR_PAD) * 12;

  if ((N % 16) == 0 && ws_size >= need) {
    char* w = (char*)d_ws;
    int* counts = (int*)w;    w += (size_t)N * 4;
    int* cursor = (int*)w;    w += (size_t)N * 4;
    int* rowStart = (int*)w;  w += (size_t)(N + 1) * 4;
    int* csrCol = (int*)w;    w += (size_t)(E + CSR_PAD) * 4;
    int* csrRow = (int*)w;    w += (size_t)(E + CSR_PAD) * 4;
    float* csrVal = (float*)w;

    const int gz = (2 * N + 255) / 256;
    const int ge = (E + 255) / 256;

    zero_i32_kernel<<<gz, 256, 0, stream>>>(counts, 2 * N);  // counts+cursor
    pad_kernel<<<1, 32, 0, stream>>>(csrCol, csrRow, csrVal, E);
    hist_kernel<<<ge, 256, 0, stream>>>(rows, counts, E);
    scan_kernel<<<1, SCAN_THREADS, 0, stream>>>(counts, rowStart, N);
    fill_kernel<<<ge, 256, 0, stream>>>(rows, cols, vals, rowStart, cursor,
                                        csrCol, csrRow, csrVal, E);
    wmma_spmm_kernel<<<N / 16, 256, 0, stream>>>(rowStart, csrCol, csrRow,
                                                 csrVal, ann, out);
  } else {
    // Fallback: gather + f32 atomic scatter (hardware global_atomic_add_f32).
    zero_f32_kernel<<<(out_size + 255) / 256, 256, 0, stream>>>(out, out_size);
    scatter_kernel<<<2048, 256, 0, stream>>>(rows, cols, vals, ann, out, E);
  }
}